// InterestExtractLayer_31396210933837
// MI455X (gfx1250) — compile-verified
//
#include <hip/hip_runtime.h>
#include <hip/hip_bf16.h>
#include <math.h>

// ---------------- problem constants ----------------
#define BB   8192
#define TT   40
#define EE   64
#define H1D  128
#define H2D  64
#define DIN  128          // 2*EE
#define TOK  (BB * (TT - 1))         // 319488 tokens for aux loss
#define NTILE_TOK (TOK / 16)         // 19968, exact

// ws layout (in floats)
#define WS_WT   0                    // gru_kernel^T   : 192 x 64
#define WS_UT   (WS_WT + 192 * 64)   // gru_rkernel^T  : 192 x 64
#define WS_W1T  (WS_UT + 192 * 64)   // w1^T           : 128 x 128
#define WS_W2T  (WS_W1T + 128 * 128) // w2^T           : 64  x 128
#define WS_ACC  (WS_W2T + 64 * 128)  // [loss_sum, mask_cnt]

typedef float v2f __attribute__((ext_vector_type(2)));
typedef float v8f __attribute__((ext_vector_type(8)));

__device__ __forceinline__ v8f wmma4(v2f a, v2f b, v8f c) {
  // D = A(16x4, f32) * B(4x16, f32) + C(16x16, f32)
  return __builtin_amdgcn_wmma_f32_16x16x4_f32(
      /*neg_a=*/false, a, /*neg_b=*/false, b,
      /*c_mod=*/(short)0, c, /*reuse_a=*/false, /*reuse_b=*/false);
}

__device__ __forceinline__ v8f bcast8(float v) {
  v8f r;
#pragma unroll
  for (int j = 0; j < 8; ++j) r[j] = v;
  return r;
}

__device__ __forceinline__ float sigmoidf_(float x) {
  return 1.0f / (1.0f + __expf(-x));
}

// K=64 MAC chain: 16 chained fp32 WMMAs. Bt is N-major [n][64].
__device__ __forceinline__ v8f mac16(const v2f* __restrict__ afrag,
                                     const float* __restrict__ Bt,
                                     int col, int lk, v8f acc) {
#pragma unroll
  for (int k = 0; k < 16; ++k) {
    v2f b = *(const v2f*)(Bt + (size_t)col * 64 + k * 4 + lk);
    acc = wmma4(afrag[k], b, acc);
  }
  return acc;
}

// K=128 MAC chain: 32 chained fp32 WMMAs. Bt is N-major [n][128].
__device__ __forceinline__ v8f mac32(const v2f* __restrict__ afrag,
                                     const float* __restrict__ Bt,
                                     int col, int lk, v8f acc) {
#pragma unroll
  for (int k = 0; k < 32; ++k) {
    v2f b = *(const v2f*)(Bt + (size_t)col * 128 + k * 4 + lk);
    acc = wmma4(afrag[k], b, acc);
  }
  return acc;
}

// ---------------- prep: transposes + accumulator zeroing ----------------
__global__ void __launch_bounds__(256, 1)
prep_kernel(const float* __restrict__ W, const float* __restrict__ U,
            const float* __restrict__ w1, const float* __restrict__ w2,
            float* __restrict__ ws) {
  int tid = threadIdx.x + blockIdx.x * blockDim.x;
  int nthr = blockDim.x * gridDim.x;
  float* Wt  = ws + WS_WT;
  float* Ut  = ws + WS_UT;
  float* w1t = ws + WS_W1T;
  float* w2t = ws + WS_W2T;
  for (int i = tid; i < 192 * 64; i += nthr) {
    int u = i >> 6, e = i & 63;
    Wt[i] = W[e * 192 + u];
    Ut[i] = U[e * 192 + u];
  }
  for (int i = tid; i < 128 * 128; i += nthr) {
    int n = i >> 7, k = i & 127;
    w1t[i] = w1[k * 128 + n];
  }
  for (int i = tid; i < 64 * 128; i += nthr) {
    int n = i >> 7, k = i & 127;
    w2t[i] = w2[k * 64 + n];
  }
  if (tid < 2) ws[WS_ACC + tid] = 0.0f;  // must re-zero every call (atomics)
}

// ---------------- GRU scan: one wave32 per 16 batch rows ----------------
// blockDim = 64 (2 waves), grid = B/16/2 = 256
__global__ void __launch_bounds__(64, 1)
gru_kernel(const float* __restrict__ ub,
           const unsigned char* __restrict__ mask,
           const float* __restrict__ bias,   // (2,192)
           const float* __restrict__ ws,
           float* __restrict__ out) {
  __shared__ float hsh[2 * 16 * 64];
  const int lane = threadIdx.x & 31;
  const int wave = threadIdx.x >> 5;
  const int rb = (blockIdx.x * 2 + wave) * 16;   // batch-row base of this tile
  float* h = hsh + wave * (16 * 64);

  const float* Wt = ws + WS_WT;   // [u=0..191][e=0..63]
  const float* Ut = ws + WS_UT;
  const float* bi = bias;
  const float* br = bias + 192;

  const int lr = lane & 15;             // A row / D column within tile
  const int hi = (lane >> 4);           // lane half
  const int lk = hi * 2;                // k sub-offset for A/B fragments

  // h0 = 0
  for (int i = lane; i < 16 * 64; i += 32) h[i] = 0.0f;

  v2f ax[16], ah[16];

  for (int t = 0; t < TT; ++t) {
    // mask for the 8 rows this lane owns in C/D layout
    float mrow[8];
#pragma unroll
    for (int j = 0; j < 8; ++j) {
      int row = rb + j + (hi << 3);
      mrow[j] = mask[(size_t)row * TT + t] ? 1.0f : 0.0f;
    }
    // preload A-fragments: x_t rows (global) and h rows (LDS).
    // Once these are in registers, LDS h may be overwritten freely.
    const float* prow = ub + ((size_t)(rb + lr) * TT + t) * EE;
#pragma unroll
    for (int k = 0; k < 16; ++k) {
      ax[k] = *(const v2f*)(prow + k * 4 + lk);
      ah[k] = *(const v2f*)(h + lr * 64 + k * 4 + lk);
    }

#pragma unroll
    for (int c = 0; c < 4; ++c) {
      const int cz = c * 16 + lr;          // z-gate column
      const int cr = 64 + c * 16 + lr;     // r-gate column
      const int ch = 128 + c * 16 + lr;    // h-gate column

      v8f accZ  = bcast8(bi[cz] + br[cz]);
      accZ  = mac16(ax, Wt, cz, lk, accZ);
      accZ  = mac16(ah, Ut, cz, lk, accZ);

      v8f accR  = bcast8(bi[cr] + br[cr]);
      accR  = mac16(ax, Wt, cr, lk, accR);
      accR  = mac16(ah, Ut, cr, lk, accR);

      v8f accXH = bcast8(bi[ch]);
      accXH = mac16(ax, Wt, ch, lk, accXH);

      v8f accRH = bcast8(br[ch]);
      accRH = mac16(ah, Ut, ch, lk, accRH);

      // previous h for this column block (C/D layout), then update + emit
#pragma unroll
      for (int j = 0; j < 8; ++j) {
        int row = j + (hi << 3);
        float hcv = h[row * 64 + c * 16 + lr];
        float z  = sigmoidf_(accZ[j]);
        float r  = sigmoidf_(accR[j]);
        float hh = tanhf(accXH[j] + r * accRH[j]);
        float nv = z * hcv + (1.0f - z) * hh;
        float v  = mrow[j] != 0.0f ? nv : hcv;
        h[row * 64 + c * 16 + lr] = v;
        out[((size_t)(rb + row) * TT + t) * EE + c * 16 + lr] = v;
      }
    }
  }
}

// ---------------- MLP + aux loss: one wave per 16 tokens ----------------
// blockDim = 128 (4 waves), grid = 19968/4 = 4992
__global__ void __launch_bounds__(128, 1)
mlp_kernel(const float* __restrict__ ub,
           const float* __restrict__ nub,
           const unsigned char* __restrict__ nmask,
           const float* __restrict__ gru,   // = d_out (B,T,E)
           const float* __restrict__ ws,
           const float* __restrict__ b1v,
           const float* __restrict__ b2v,
           const float* __restrict__ w3v,
           const float* __restrict__ b3v,
           float* __restrict__ accum) {
  __shared__ float s1all[4 * 16 * 128];   // per-wave 16x128 staging (32 KB)
  const int lane = threadIdx.x & 31;
  const int wave = threadIdx.x >> 5;
  const int tile = blockIdx.x * 4 + wave;
  float* sc = s1all + wave * (16 * 128);

  const float* w1t = ws + WS_W1T;  // [n=0..127][k=0..127]
  const float* w2t = ws + WS_W2T;  // [n=0..63][k=0..127]

  const int lr = lane & 15;
  const int hi = lane >> 4;
  const int lk = hi * 2;

  // this lane's token (row lr of the tile)
  const int tau = tile * 16 + lr;
  const int b   = tau / (TT - 1);
  const int tp  = tau % (TT - 1);
  const float* pEmb = gru + ((size_t)b * TT + tp + 1) * EE;   // gru_embed row
  const float* pPos = ub  + ((size_t)b * TT + tp + 1) * EE;   // behavior row
  const float* pNeg = nub + ((size_t)b * (TT - 1) + tp) * EE; // neg row
  const float  m    = nmask[(size_t)b * (TT - 1) + tp] ? 1.0f : 0.0f;

  const float LOG_E   = -16.118095651f;      // log(1e-7)
  const float LOG_1ME = -1.00000005e-7f;     // log(1 - 1e-7)

  float lsum = 0.0f;
  v2f a[32];

  for (int path = 0; path < 2; ++path) {
    const float* pSecond = path ? pNeg : pPos;
    // A-fragments for K=128: k-steps 0..15 read gru_embed, 16..31 read second half
#pragma unroll
    for (int k = 0; k < 16; ++k)
      a[k] = *(const v2f*)(pEmb + k * 4 + lk);
#pragma unroll
    for (int k = 16; k < 32; ++k)
      a[k] = *(const v2f*)(pSecond + (k - 16) * 4 + lk);

    // GEMM1: 16x128 @ 128x128 -> relu -> LDS
#pragma unroll
    for (int ct = 0; ct < 8; ++ct) {
      const int col = ct * 16 + lr;
      v8f acc = bcast8(b1v[col]);
      acc = mac32(a, w1t, col, lk, acc);
#pragma unroll
      for (int j = 0; j < 8; ++j) {
        float v = acc[j] > 0.0f ? acc[j] : 0.0f;
        sc[(j + (hi << 3)) * 128 + col] = v;
      }
    }
    // GEMM2: 16x128 @ 128x64 -> relu
#pragma unroll
    for (int k = 0; k < 32; ++k)
      a[k] = *(const v2f*)(sc + lr * 128 + k * 4 + lk);
    v8f h2[4];
#pragma unroll
    for (int ct = 0; ct < 4; ++ct) {
      const int col = ct * 16 + lr;
      v8f acc = bcast8(b2v[col]);
      acc = mac32(a, w2t, col, lk, acc);
#pragma unroll
      for (int j = 0; j < 8; ++j) acc[j] = acc[j] > 0.0f ? acc[j] : 0.0f;
      h2[ct] = acc;
    }
    // stage h2 (16x64) into LDS, then final 64-wide dot on lanes 0..15
#pragma unroll
    for (int ct = 0; ct < 4; ++ct)
#pragma unroll
      for (int j = 0; j < 8; ++j)
        sc[(j + (hi << 3)) * 64 + ct * 16 + lr] = h2[ct][j];

    float lg = b3v[0];
    if (lane < 16) {
#pragma unroll 8
      for (int e = 0; e < 64; ++e) lg += sc[lr * 64 + e] * w3v[e];
    }
    float p = sigmoidf_(lg);
    float lterm = (path == 0)
                      ? -(p * LOG_1ME + (1.0f - p) * LOG_E)     // l_pos
                      : -(p * LOG_E + (1.0f - p) * LOG_1ME);    // l_neg
    lsum += lterm;
  }

  float contrib = (lane < 16) ? lsum * m : 0.0f;
  float cnt     = (lane < 16) ? m : 0.0f;
#pragma unroll
  for (int off = 16; off > 0; off >>= 1) {
    contrib += __shfl_xor(contrib, off, 32);
    cnt     += __shfl_xor(cnt, off, 32);
  }
  if (lane == 0) {
    atomicAdd(&accum[0], contrib);
    atomicAdd(&accum[1], cnt);
  }
}

// ---------------- finalize ----------------
__global__ void __launch_bounds__(32, 1)
finalize_kernel(const float* __restrict__ accum,
                float* __restrict__ out_loss) {
  if (threadIdx.x != 0) return;
  float cnt = accum[1];
  if (cnt < 1.0f) cnt = 1.0f;
  *out_loss = accum[0] / (2.0f * cnt);
}

extern "C" void kernel_launch(void* const* d_in, const int* in_sizes, int n_in,
                              void* d_out, int out_size, void* d_ws, size_t ws_size,
                              hipStream_t stream) {
  const float*         ub    = (const float*)d_in[0];
  const unsigned char* mask  = (const unsigned char*)d_in[1];
  const float*         nub   = (const float*)d_in[2];
  const unsigned char* nmask = (const unsigned char*)d_in[3];
  const float*         W     = (const float*)d_in[4];
  const float*         U     = (const float*)d_in[5];
  const float*         bias  = (const float*)d_in[6];
  const float*         w1    = (const float*)d_in[7];
  const float*         b1v   = (const float*)d_in[8];
  const float*         w2    = (const float*)d_in[9];
  const float*         b2v   = (const float*)d_in[10];
  const float*         w3v   = (const float*)d_in[11];
  const float*         b3v   = (const float*)d_in[12];
  float* out = (float*)d_out;
  float* ws  = (float*)d_ws;

  prep_kernel<<<64, 256, 0, stream>>>(W, U, w1, w2, ws);
  gru_kernel<<<BB / 16 / 2, 64, 0, stream>>>(ub, mask, bias, ws, out);
  mlp_kernel<<<NTILE_TOK / 4, 128, 0, stream>>>(ub, nub, nmask, out, ws,
                                                b1v, b2v, w3v, b3v, ws + WS_ACC);
  finalize_kernel<<<1, 32, 0, stream>>>(ws + WS_ACC, out + (size_t)BB * TT * EE);
}